// MultiboxLoss_26577257628029
// MI455X (gfx1250) — compile-verified
//
#include <hip/hip_runtime.h>
#include <hip/hip_bf16.h>

constexpr int NB = 64;      // batch
constexpr int NP = 16384;   // priors
constexpr int NC = 81;      // classes
constexpr int NK = 32;      // truths per image
constexpr int GX = NP / 256; // 64 blocks along prior dim

typedef float v2f __attribute__((ext_vector_type(2)));
typedef float v8f __attribute__((ext_vector_type(8)));

// ---------------------------------------------------------------------------
// WMMA block reduction: sum 256 floats staged in LDS using V_WMMA_F32_16X16X4_F32
// with a ones B-matrix. D[m,n] = sum_k A[m,k] -> row sums of the 16x16 tile,
// accumulated across 4 K-chunks. Full f32 precision. EXEC must be all ones
// (call under wave-uniform branch).
// A 16x4 f32 layout: lanes 0-15: M=lane, VGPR0=K0,VGPR1=K1; lanes 16-31: K2,K3.
// D layout: VGPR r: lanes 0-15 -> M=r, lanes 16-31 -> M=8+r (replicated over N).
// ---------------------------------------------------------------------------
__device__ inline float wmma_sum256(const float* s, int lane) {
    int half = lane >> 4;     // 0: K{0,1}, 1: K{2,3}
    int m    = lane & 15;     // matrix row
    v8f c = {0.f, 0.f, 0.f, 0.f, 0.f, 0.f, 0.f, 0.f};
    v2f ones; ones.x = 1.0f; ones.y = 1.0f;
#pragma unroll
    for (int j = 0; j < 4; ++j) {
        v2f a;
        int k0 = 4 * j + 2 * half;
        a.x = s[m * 16 + k0];
        a.y = s[m * 16 + k0 + 1];
        c = __builtin_amdgcn_wmma_f32_16x16x4_f32(
            false, a, false, ones, (short)0, c, false, false);
    }
    // lanes 0-15 hold row sums of rows 0..7 in c[0..7]; lanes 16-31 rows 8..15
    float r = c[0] + c[1] + c[2] + c[3] + c[4] + c[5] + c[6] + c[7];
    float other = __shfl(r, lane ^ 16, 32);
    return r + other;
}

// ---------------------------------------------------------------------------
// Kernel 1: SSD matching per batch image.
// ---------------------------------------------------------------------------
__global__ __launch_bounds__(256) void match_kernel(
    const float* __restrict__ targets,  // [B,K,4] corner
    const float* __restrict__ priors,   // [P,4] center
    float* __restrict__ ov,             // [B,P] best truth overlap
    int* __restrict__ tix)              // [B,P] best truth idx
{
    int b = blockIdx.x, tid = threadIdx.x;
    __shared__ float4 tr[NK];
    __shared__ float  tarea[NK];
    __shared__ float  sv[8 * 256];
    __shared__ unsigned short sp[8 * 256];
    __shared__ int bestPrior[NK];

    if (tid < NK) {
        float4 t = ((const float4*)targets)[b * NK + tid];
        tr[tid] = t;
        tarea[tid] = (t.z - t.x) * (t.w - t.y);
    }
    __syncthreads();

    // Pass A: per-prior best truth (first-max tie-break => strict >, k ascending)
    for (int i = 0; i < GX; ++i) {
        int p = tid + i * 256;
        float4 pr = ((const float4*)priors)[p];
        float x1 = pr.x - pr.z * 0.5f, y1 = pr.y - pr.w * 0.5f;
        float x2 = pr.x + pr.z * 0.5f, y2 = pr.y + pr.w * 0.5f;
        float pa = (x2 - x1) * (y2 - y1);
        float bv = -1.0f; int bk = 0;
        for (int k = 0; k < NK; ++k) {
            float4 t = tr[k];
            float iw = fmaxf(fminf(x2, t.z) - fmaxf(x1, t.x), 0.f);
            float ih = fmaxf(fminf(y2, t.w) - fmaxf(y1, t.y), 0.f);
            float inter = iw * ih;
            float iou = inter / (pa + tarea[k] - inter);
            if (iou > bv) { bv = iou; bk = k; }
        }
        ov[(size_t)b * NP + p]  = bv;
        tix[(size_t)b * NP + p] = bk;
    }

    // Pass B: per-truth best prior, chunks of 8 truths (register-local argmax)
    for (int kc = 0; kc < NK; kc += 8) {
        float bv[8]; int bp[8];
#pragma unroll
        for (int j = 0; j < 8; ++j) { bv[j] = -1.0f; bp[j] = 0; }
        for (int i = 0; i < GX; ++i) {
            int p = tid + i * 256;
            float4 pr = ((const float4*)priors)[p];
            float x1 = pr.x - pr.z * 0.5f, y1 = pr.y - pr.w * 0.5f;
            float x2 = pr.x + pr.z * 0.5f, y2 = pr.y + pr.w * 0.5f;
            float pa = (x2 - x1) * (y2 - y1);
#pragma unroll
            for (int j = 0; j < 8; ++j) {
                float4 t = tr[kc + j];
                float iw = fmaxf(fminf(x2, t.z) - fmaxf(x1, t.x), 0.f);
                float ih = fmaxf(fminf(y2, t.w) - fmaxf(y1, t.y), 0.f);
                float inter = iw * ih;
                float iou = inter / (pa + tarea[kc + j] - inter);
                if (iou > bv[j]) { bv[j] = iou; bp[j] = p; }  // p ascending per thread
            }
        }
#pragma unroll
        for (int j = 0; j < 8; ++j) {
            sv[j * 256 + tid] = bv[j];
            sp[j * 256 + tid] = (unsigned short)bp[j];
        }
        __syncthreads();
        if (tid < 8) {
            float best = -1.0f; int bpi = NP;
            for (int t2 = 0; t2 < 256; ++t2) {
                float v = sv[tid * 256 + t2];
                int pi = (int)sp[tid * 256 + t2];
                if (v > best || (v == best && pi < bpi)) { best = v; bpi = pi; }
            }
            bestPrior[kc + tid] = bpi;
        }
        __syncthreads();
    }

    // Sequential overrides (matches .at[].set(2.0) + fori last-write-wins)
    if (tid == 0) {
        for (int j = 0; j < NK; ++j) {
            int p = bestPrior[j];
            ov[(size_t)b * NP + p]  = 2.0f;
            tix[(size_t)b * NP + p] = j;
        }
    }
}

// ---------------------------------------------------------------------------
// Kernel 2: per-(b,p) decode+GIoU, one-pass 81-class CE, mining values,
// WMMA block reductions into deterministic per-block partials.
// ---------------------------------------------------------------------------
__global__ __launch_bounds__(256) void main_kernel(
    const float* __restrict__ conf,     // [B,P,C]
    const float* __restrict__ loc,      // [B,P,4]
    const float* __restrict__ targets,  // [B,K,4]
    const int*   __restrict__ labels,   // [B,K]
    const float* __restrict__ priors,   // [P,4]
    const float* __restrict__ ov,
    const int*   __restrict__ tix,
    float* __restrict__ mine,           // [B,P]
    float* __restrict__ pg,             // [B*GX] giou partials
    float* __restrict__ pce,            // [B*GX] positive-CE partials
    float* __restrict__ pnp)            // [B*GX] positive-count partials
{
    __shared__ float red[3][256];
    int b = blockIdx.y, tid = threadIdx.x;
    int p = blockIdx.x * 256 + tid;
    size_t bp = (size_t)b * NP + p;

    float o  = ov[bp];
    int   ti = tix[bp];
    int   ct = (o < 0.5f) ? 0 : labels[b * NK + ti];
    bool pos = ct > 0;

    // ---- localization: decode + GIoU vs matched truth
    float4 pr = ((const float4*)priors)[p];
    float4 ld = ((const float4*)loc)[bp];
    float cx = pr.x + ld.x * 0.1f * pr.z;
    float cy = pr.y + ld.y * 0.1f * pr.w;
    float w  = pr.z * expf(ld.z * 0.2f);
    float h  = pr.w * expf(ld.w * 0.2f);
    float ax1 = cx - w * 0.5f, ay1 = cy - h * 0.5f;
    float ax2 = cx + w * 0.5f, ay2 = cy + h * 0.5f;
    float4 tb = ((const float4*)targets)[b * NK + ti];
    float iw = fmaxf(fminf(ax2, tb.z) - fmaxf(ax1, tb.x), 0.f);
    float ih = fmaxf(fminf(ay2, tb.w) - fmaxf(ay1, tb.y), 0.f);
    float inter  = iw * ih;
    float area_a = (ax2 - ax1) * (ay2 - ay1);
    float area_b = (tb.z - tb.x) * (tb.w - tb.y);
    float uni = area_a + area_b - inter;
    float ew = fmaxf(fmaxf(ax2, tb.z) - fminf(ax1, tb.x), 0.f);
    float eh = fmaxf(fmaxf(ay2, tb.w) - fminf(ay1, tb.y), 0.f);
    float enc = ew * eh;
    float g = 1.0f - (inter / uni - (enc - uni) / enc);

    // ---- confidence: CE = logsumexp(row) - row[ct], single global pass
    const float* row = conf + bp * (size_t)NC;
    float x[NC];
#pragma unroll
    for (int c = 0; c < NC; ++c) x[c] = row[c];
    float m = x[0];
#pragma unroll
    for (int c = 1; c < NC; ++c) m = fmaxf(m, x[c]);
    float s = 0.0f;
    float gat = x[0];
#pragma unroll
    for (int c = 0; c < NC; ++c) {
        s += expf(x[c] - m);
        if (c > 0 && c == ct) gat = x[c];
    }
    float ce = fmaxf(m + logf(s) - gat, 0.0f);   // clamp to +0: keeps float bits orderable

    mine[bp] = pos ? 0.0f : ce;

    red[0][tid] = pos ? g : 0.0f;
    red[1][tid] = pos ? ce : 0.0f;
    red[2][tid] = pos ? 1.0f : 0.0f;
    __syncthreads();

    int wid = tid >> 5, lane = tid & 31;
    if (wid < 3) {                      // wave-uniform: EXEC all ones inside
        float tot = wmma_sum256(&red[wid][0], lane);
        if (lane == 0) {
            int pb = b * GX + blockIdx.x;
            if (wid == 0)      pg[pb]  = tot;
            else if (wid == 1) pce[pb] = tot;
            else               pnp[pb] = tot;
        }
    }
}

// ---------------------------------------------------------------------------
// Kernel 3: hard negative mining = sum of top-M mine values per row.
// 4-pass radix select (8-bit digits, MSB first) over non-negative float bits,
// then threshold sum:  S = sum_{v>t} v + need*t.
// ---------------------------------------------------------------------------
__global__ __launch_bounds__(256) void mine_kernel(
    const float* __restrict__ mine,
    const float* __restrict__ pnp,
    float* __restrict__ negrow)
{
    int b = blockIdx.x, tid = threadIdx.x;
    __shared__ unsigned hist[256];
    __shared__ float s_red[256];
    __shared__ int s_need;
    __shared__ unsigned s_prefix;
    __shared__ int s_np;

    if (tid == 0) {
        float np = 0.0f;
        for (int i = 0; i < GX; ++i) np += pnp[b * GX + i];  // exact (small ints)
        s_np = (int)np;
    }
    __syncthreads();
    int M = min(3 * s_np, NP - 1);
    const float* row = mine + (size_t)b * NP;

    unsigned prefix = 0;
    int need = M;
    for (int pass = 0; pass < 4; ++pass) {
        int shift = 24 - 8 * pass;
        hist[tid] = 0;
        __syncthreads();
        for (int i = 0; i < GX; ++i) {
            unsigned u = __float_as_uint(row[tid + i * 256]);
            bool match = (pass == 0) || ((u >> (shift + 8)) == prefix);
            if (match) atomicAdd(&hist[(u >> shift) & 255u], 1u);
        }
        __syncthreads();
        if (tid == 0) {
            int cum = 0, d = 255;
            for (; d > 0; --d) {
                int c = (int)hist[d];
                if (cum + c >= need) break;
                cum += c;
            }
            s_prefix = (prefix << 8) | (unsigned)d;
            s_need = need - cum;
        }
        __syncthreads();
        prefix = s_prefix;
        need = s_need;
    }

    unsigned tb = prefix;
    float t = __uint_as_float(prefix);
    float local = 0.0f;
    for (int i = 0; i < GX; ++i) {
        float v = row[tid + i * 256];
        if (__float_as_uint(v) > tb) local += v;
    }
    s_red[tid] = local;
    __syncthreads();
    if (tid == 0) {
        float tot = 0.0f;
        for (int i = 0; i < 256; ++i) tot += s_red[i];
        negrow[b] = (M > 0) ? (tot + (float)need * t) : 0.0f;
    }
}

// ---------------------------------------------------------------------------
// Kernel 4: deterministic serial finalize.
// ---------------------------------------------------------------------------
__global__ void finalize_kernel(
    const float* __restrict__ pg, const float* __restrict__ pce,
    const float* __restrict__ pnp, const float* __restrict__ negrow,
    float* __restrict__ out)
{
    if (threadIdx.x == 0 && blockIdx.x == 0) {
        float sg = 0.f, sce = 0.f, N = 0.f, neg = 0.f;
        for (int i = 0; i < NB * GX; ++i) { sg += pg[i]; sce += pce[i]; N += pnp[i]; }
        for (int b = 0; b < NB; ++b) neg += negrow[b];
        out[0] = 5.0f * sg / N;
        out[1] = (sce + neg) / N;
    }
}

// ---------------------------------------------------------------------------
extern "C" void kernel_launch(void* const* d_in, const int* in_sizes, int n_in,
                              void* d_out, int out_size, void* d_ws, size_t ws_size,
                              hipStream_t stream) {
    const float* conf    = (const float*)d_in[0];
    const float* loc     = (const float*)d_in[1];
    const float* targets = (const float*)d_in[2];
    const int*   labels  = (const int*)d_in[3];
    const float* priors  = (const float*)d_in[4];
    float* out = (float*)d_out;

    float* wsf    = (float*)d_ws;
    float* pg     = wsf;                       // [B*GX]
    float* pce    = wsf + NB * GX;             // [B*GX]
    float* pnp    = wsf + 2 * NB * GX;         // [B*GX]
    float* negrow = wsf + 3 * NB * GX;         // [B]
    float* ov     = wsf + 16384;               // [B*P]
    int*   tix    = (int*)(ov + (size_t)NB * NP);
    float* mine   = (float*)(tix + (size_t)NB * NP);

    match_kernel<<<NB, 256, 0, stream>>>(targets, priors, ov, tix);
    main_kernel<<<dim3(GX, NB), 256, 0, stream>>>(conf, loc, targets, labels,
                                                  priors, ov, tix, mine, pg, pce, pnp);
    mine_kernel<<<NB, 256, 0, stream>>>(mine, pnp, negrow);
    finalize_kernel<<<1, 1, 0, stream>>>(pg, pce, pnp, negrow, out);
}